// LogSplineFilterFullVI_37675453120790
// MI455X (gfx1250) — compile-verified
//
#include <hip/hip_runtime.h>
#include <math.h>

#define NSPAN 9
#define JDIM  12

// ---------------------------------------------------------------------------
// Compile-time knot machinery.  Knots: [T_MIN x4, T_MIN+h..T_MIN+8h, T_MAX x4]
// T_MIN = ln(1e-6), T_MAX = ln(2 + 1e-6)  (double precision constants)
// ---------------------------------------------------------------------------
static constexpr double T_MIN_D = -13.815510557964274;
static constexpr double T_MAX_D =  0.69314768055982031;
static constexpr double H_D     = (T_MAX_D - T_MIN_D) / (double)NSPAN;

constexpr double knotv(int i) {
  int j = i - 3;
  if (j < 0) j = 0;
  if (j > NSPAN) j = NSPAN;
  return T_MIN_D + (double)j * H_D;
}

struct P4 { double c[4]; };

// Symbolic Cox-de Boor: cubic basis function B_j^3 restricted to span sp,
// expressed as a polynomial in u = t - knot(3+sp).  Pure constexpr.
constexpr P4 basis_poly(int j, int sp) {
  double B[4][4] = {};
  for (int i = 0; i < 4; ++i)
    B[i][0] = ((j + i) == (3 + sp)) ? 1.0 : 0.0;
  for (int k = 1; k <= 3; ++k) {
    for (int ii = 0; ii <= 3 - k; ++ii) {
      const int i = j + ii;
      const double d1 = knotv(i + k) - knotv(i);
      const double d2 = knotv(i + k + 1) - knotv(i + 1);
      double a0 = 0.0, a1 = 0.0, b0 = 0.0, b1 = 0.0;
      if (d1 != 0.0) { a0 = (knotv(3 + sp) - knotv(i)) / d1;        a1 =  1.0 / d1; }
      if (d2 != 0.0) { b0 = (knotv(i + k + 1) - knotv(3 + sp)) / d2; b1 = -1.0 / d2; }
      double r[4] = {};
      for (int c = 0; c < 4; ++c) {
        r[c] += a0 * B[ii][c] + b0 * B[ii + 1][c];
        if (c + 1 < 4) r[c + 1] += a1 * B[ii][c] + b1 * B[ii + 1][c];
      }
      for (int c = 0; c < 4; ++c) B[ii][c] = r[c];
    }
  }
  return P4{ { B[0][0], B[0][1], B[0][2], B[0][3] } };
}

struct SplineTab { float P[NSPAN][4][4]; };  // [span][local basis][poly coeff]

constexpr SplineTab make_tab() {
  SplineTab t = {};
  for (int sp = 0; sp < NSPAN; ++sp)
    for (int i = 0; i < 4; ++i) {
      P4 p = basis_poly(sp + i, sp);
      for (int c = 0; c < 4; ++c) t.P[sp][i][c] = (float)p.c[c];
    }
  return t;
}

static __constant__ SplineTab g_tab = make_tab();

// ---------------------------------------------------------------------------
// Setup kernel (1 wave of 32): w = mu_w + L_w @ eps_w via V_WMMA_F32_16X16X4_F32,
// then fold w into per-span cubic coefficients + scalar C = exp(log_tau2 - s0).
// d_ws layout (floats): [0..35] = tbl[9][4], [36] = C.
// ---------------------------------------------------------------------------
typedef __attribute__((ext_vector_type(2))) float v2f;
typedef __attribute__((ext_vector_type(8))) float v8f;

__global__ __launch_bounds__(32) void spline_setup_kernel(
    const float* __restrict__ mu_lt2, const float* __restrict__ lstd,
    const float* __restrict__ mu_w,   const float* __restrict__ Lw,
    const float* __restrict__ e_tau,  const float* __restrict__ e_w,
    float* __restrict__ wsf)
{
  const int lane = (int)threadIdx.x;
  const int m    = lane & 15;     // A-matrix row
  const int half = lane >> 4;     // K half select (A layout: lanes 0-15 K=0,1; 16-31 K=2,3)

  v8f acc = {};
#pragma unroll
  for (int kc = 0; kc < 4; ++kc) {          // K = 16 total, 4 per WMMA
    const int k0 = kc * 4 + half * 2;
    // A fragment: L_w zero-padded to 16x16 (branchless: clamp index, mask value)
    const float am0 = (m < JDIM && (k0    ) < JDIM) ? 1.f : 0.f;
    const float am1 = (m < JDIM && (k0 + 1) < JDIM) ? 1.f : 0.f;
    const int ia0 = (m * JDIM + k0     < JDIM * JDIM) ? m * JDIM + k0     : JDIM * JDIM - 1;
    const int ia1 = (m * JDIM + k0 + 1 < JDIM * JDIM) ? m * JDIM + k0 + 1 : JDIM * JDIM - 1;
    v2f a; a.x = Lw[ia0] * am0; a.y = Lw[ia1] * am1;
    // B fragment: row k of B = eps_w[k] broadcast across all 16 columns,
    // so D[:,n] is identical for all n (column-invariant => N-layout irrelevant)
    const float bm0 = ((k0    ) < JDIM) ? 1.f : 0.f;
    const float bm1 = ((k0 + 1) < JDIM) ? 1.f : 0.f;
    const int ib0 = (k0     < JDIM) ? k0     : JDIM - 1;
    const int ib1 = (k0 + 1 < JDIM) ? k0 + 1 : JDIM - 1;
    v2f b; b.x = e_w[ib0] * bm0; b.y = e_w[ib1] * bm1;
    acc = __builtin_amdgcn_wmma_f32_16x16x4_f32(
        /*neg_a=*/false, a, /*neg_b=*/false, b,
        /*c_mod=*/(short)0, acc, /*reuse_a=*/false, /*reuse_b=*/false);
  }

  // C/D layout: lane 0 column 0 holds rows 0..7 in acc[0..7];
  //             lane 16 column 0 holds rows 8..15 in acc[0..7].
  __shared__ float wsh[16];
  if (lane == 0) {
#pragma unroll
    for (int r = 0; r < 8; ++r) wsh[r] = mu_w[r] + acc[r];
  }
  if (lane == 16) {
#pragma unroll
    for (int r = 0; r < 4; ++r) wsh[8 + r] = mu_w[8 + r] + acc[r];
  }
  __syncthreads();

  // tbl[sp][k] = sum_i w[sp+i] * P[sp][i][k]   (36 entries, 32 lanes)
  for (int e = lane; e < NSPAN * 4; e += 32) {
    const int sp = e >> 2, k = e & 3;
    float t = 0.f;
#pragma unroll
    for (int i = 0; i < 4; ++i)
      t = __builtin_fmaf(wsh[sp + i], g_tab.P[sp][i][k], t);
    wsf[e] = t;
    if (e == 0) {  // s0 = s(T_MIN) = c0 of span 0 == this entry
      const float lt2 = mu_lt2[0] + __expf(lstd[0]) * e_tau[0];
      wsf[36] = __expf(lt2 - t);
    }
  }
}

// ---------------------------------------------------------------------------
// Main kernel: memory-bound elementwise spline eval.
// Per element: v_log_f32, 1 fma + clamp for span, ds_load_b128 of 4 coeffs,
// 3-FMA Horner, v_exp_f32, 1 mul.  float4 I/O -> b128 global ops.
// ---------------------------------------------------------------------------
static constexpr float T_MINF = (float)T_MIN_D;
static constexpr float HF     = (float)H_D;
static constexpr float INV_HF = (float)(1.0 / H_D);

__global__ __launch_bounds__(256) void spline_eval_kernel(
    const float* __restrict__ lam, const float* __restrict__ wsf,
    float* __restrict__ out, int n)
{
  __shared__ __align__(16) float tbl[NSPAN * 4];
  const int tid = (int)threadIdx.x;
  if (tid < NSPAN * 4) tbl[tid] = wsf[tid];
  const float scaleC = wsf[36];   // uniform -> scalar load
  __syncthreads();

  const int base = ((int)blockIdx.x * 256 + tid) * 4;
  if (base + 3 < n) {
    const float4 v = *reinterpret_cast<const float4*>(lam + base);
    const float xin[4] = { v.x, v.y, v.z, v.w };
    float r[4];
#pragma unroll
    for (int e = 0; e < 4; ++e) {
      const float t  = __logf(xin[e] + 1e-6f);
      const float fx = (t - T_MINF) * INV_HF;
      int sp = (int)fx;
      sp = sp < 0 ? 0 : (sp > NSPAN - 1 ? NSPAN - 1 : sp);
      const float u = t - __builtin_fmaf((float)sp, HF, T_MINF);
      const float4 c = *reinterpret_cast<const float4*>(&tbl[sp * 4]);  // ds_load_b128
      const float s = __builtin_fmaf(
          __builtin_fmaf(__builtin_fmaf(c.w, u, c.z), u, c.y), u, c.x);
      r[e] = scaleC * __expf(s);
    }
    *reinterpret_cast<float4*>(out + base) = make_float4(r[0], r[1], r[2], r[3]);
  } else {
    for (int e = 0; e < 4; ++e) {
      const int idx = base + e;
      if (idx >= n) break;
      const float t  = __logf(lam[idx] + 1e-6f);
      const float fx = (t - T_MINF) * INV_HF;
      int sp = (int)fx;
      sp = sp < 0 ? 0 : (sp > NSPAN - 1 ? NSPAN - 1 : sp);
      const float u = t - __builtin_fmaf((float)sp, HF, T_MINF);
      const float4 c = *reinterpret_cast<const float4*>(&tbl[sp * 4]);
      const float s = __builtin_fmaf(
          __builtin_fmaf(__builtin_fmaf(c.w, u, c.z), u, c.y), u, c.x);
      out[idx] = scaleC * __expf(s);
    }
  }
}

// ---------------------------------------------------------------------------
extern "C" void kernel_launch(void* const* d_in, const int* in_sizes, int n_in,
                              void* d_out, int out_size, void* d_ws, size_t ws_size,
                              hipStream_t stream)
{
  const float* lam    = (const float*)d_in[0];
  const float* mu_lt2 = (const float*)d_in[1];
  const float* lstd   = (const float*)d_in[2];
  const float* mu_w   = (const float*)d_in[3];
  const float* Lw     = (const float*)d_in[4];
  const float* e_tau  = (const float*)d_in[5];
  const float* e_w    = (const float*)d_in[6];
  float* outp = (float*)d_out;
  float* wsf  = (float*)d_ws;
  const int n = in_sizes[0];

  spline_setup_kernel<<<1, 32, 0, stream>>>(mu_lt2, lstd, mu_w, Lw, e_tau, e_w, wsf);

  const int threads = 256;                    // 8 waves (wave32)
  const int epb = threads * 4;                // float4 per thread
  const int blocks = (n + epb - 1) / epb;     // 4096 blocks for N=4M
  spline_eval_kernel<<<blocks, threads, 0, stream>>>(lam, wsf, outp, n);
}